// RGCNSparseTIRTensorCoresLayer_58411555226292
// MI455X (gfx1250) — compile-verified
//
#include <hip/hip_runtime.h>
#include <stdint.h>
#include <stddef.h>

typedef __attribute__((ext_vector_type(2))) float v2f;
typedef __attribute__((ext_vector_type(4))) float v4f;
typedef __attribute__((ext_vector_type(8))) float v8f;

#define F_DIM 64  // F_IN == F_OUT == 64 per reference

// ---------------------------------------------------------------------------
// Native f32 global atomic add (non-returning), device scope.
// Avoids any CAS-loop lowering of atomicAdd(float*) — we want the CDNA5
// GLOBAL_ATOMIC_ADD_F32 path straight to L2 atomic units.
// ---------------------------------------------------------------------------
__device__ __forceinline__ void gadd_f32(float* p, float v) {
    asm volatile("global_atomic_add_f32 %0, %1, off scope:SCOPE_DEV"
                 :: "v"((uint64_t)(uintptr_t)p), "v"(v)
                 : "memory");
}

// ---------------------------------------------------------------------------
// Zero-fill d_out (harness poisons it with 0xAA before timing).
// ---------------------------------------------------------------------------
__global__ __launch_bounds__(256) void zero_kernel(float* __restrict__ p, long n) {
    long tid    = (long)blockIdx.x * blockDim.x + threadIdx.x;
    long stride = (long)gridDim.x * blockDim.x;
    long n4     = n >> 2;
    v4f z = {0.0f, 0.0f, 0.0f, 0.0f};
    for (long j = tid; j < n4; j += stride) ((v4f*)p)[j] = z;
    if (tid == 0) {
        for (long j = n4 << 2; j < n; ++j) p[j] = 0.0f;
    }
}

// ---------------------------------------------------------------------------
// Kernel 1: XW[r, n, :] = X[n, :] @ W[r, :, :] via V_WMMA_F32_16X16X4_F32.
// One wave (32 lanes) per 16-row tile of X. A-fragments for the whole K=64
// strip are preloaded once (16 x v2f = 32 VGPRs) and reused across all 8
// relations, so X is streamed from HBM exactly once.
//
// A 16x4 f32 layout: lane<16: M=lane,   VGPR0->K=0, VGPR1->K=1
//                    lane>=16: M=lane-16, VGPR0->K=2, VGPR1->K=3
// B 4x16 f32 layout mirrors A (K rows striped the same way).
// C/D 16x16 f32: VGPR v holds M = v + 8*(lane>=16), N = lane&15.
// ---------------------------------------------------------------------------
__global__ __launch_bounds__(32)
void rgcn_xw_wmma(const float* __restrict__ X,
                  const float* __restrict__ W,
                  float* __restrict__ XW,
                  int nNodes, int nRels) {
    const int lane    = threadIdx.x;      // 0..31, one wave per block
    const int half    = lane >> 4;        // 0 or 1
    const int l15     = lane & 15;
    const int rowBase = blockIdx.x * 16;

    // Preload A fragments for all 16 K-steps (K = 16 steps of 4).
    int arow = rowBase + l15;
    int arowc = arow < nNodes ? arow : (nNodes - 1);   // clamp (stores guarded)
    const float* xrow = X + (size_t)arowc * F_DIM + half * 2;
    v2f afrag[16];
#pragma unroll
    for (int k = 0; k < 16; ++k)
        afrag[k] = *(const v2f*)(xrow + k * 4);

    for (int r = 0; r < nRels; ++r) {
        const float* Wr = W + (size_t)r * F_DIM * F_DIM;
        v8f c0 = {}, c1 = {}, c2 = {}, c3 = {};
#pragma unroll
        for (int k = 0; k < 16; ++k) {
            // B fragment row base for this K-step: rows (k*4 + half*2, +1)
            const float* wrow = Wr + (size_t)(k * 4 + half * 2) * F_DIM + l15;
            v2f b0, b1, b2, b3;
            b0.x = wrow[0];        b0.y = wrow[0 + F_DIM];
            b1.x = wrow[16];       b1.y = wrow[16 + F_DIM];
            b2.x = wrow[32];       b2.y = wrow[32 + F_DIM];
            b3.x = wrow[48];       b3.y = wrow[48 + F_DIM];
            c0 = __builtin_amdgcn_wmma_f32_16x16x4_f32(false, afrag[k], false, b0, (short)0, c0, false, false);
            c1 = __builtin_amdgcn_wmma_f32_16x16x4_f32(false, afrag[k], false, b1, (short)0, c1, false, false);
            c2 = __builtin_amdgcn_wmma_f32_16x16x4_f32(false, afrag[k], false, b2, (short)0, c2, false, false);
            c3 = __builtin_amdgcn_wmma_f32_16x16x4_f32(false, afrag[k], false, b3, (short)0, c3, false, false);
        }
        // Store 16x64 tile: row = rowBase + v + 8*half, col = nt*16 + l15.
        float* out = XW + ((size_t)r * nNodes + rowBase) * F_DIM;
#pragma unroll
        for (int v = 0; v < 8; ++v) {
            int row = v + half * 8;
            if (rowBase + row < nNodes) {
                float* orow = out + (size_t)row * F_DIM + l15;
                orow[0]  = c0[v];
                orow[16] = c1[v];
                orow[32] = c2[v];
                orow[48] = c3[v];
            }
        }
    }
}

// ---------------------------------------------------------------------------
// Kernel 2: per-edge gather + scale + scatter-add.
// One wave per edge; lane l handles output features {2l, 2l+1} (coalesced
// float2 message load, two f32 atomics into Y[dst]).
// ---------------------------------------------------------------------------
__global__ __launch_bounds__(256)
void rgcn_edge_scatter(const float* __restrict__ XW,
                       const float* __restrict__ A,
                       const int* __restrict__ src,
                       const int* __restrict__ dst,
                       const int* __restrict__ etype,
                       float* __restrict__ Y,
                       int nEdges, int nNodes) {
    const int lane = threadIdx.x & 31;
    const long wave0  = (long)blockIdx.x * (blockDim.x >> 5) + (threadIdx.x >> 5);
    const long nWaves = (long)gridDim.x * (blockDim.x >> 5);
    for (long e = wave0; e < nEdges; e += nWaves) {
        int   s  = src[e];
        int   d  = dst[e];
        int   et = etype[e];
        float w  = A[e];
        const float* m = XW + ((size_t)et * nNodes + s) * F_DIM + lane * 2;
        v2f mv = *(const v2f*)m;
        float* y = Y + (size_t)d * F_DIM + lane * 2;
        gadd_f32(y + 0, mv.x * w);
        gadd_f32(y + 1, mv.y * w);
    }
}

// ---------------------------------------------------------------------------
// Fallback (only if ws_size can't hold XW): fused per-edge GEMV + scatter.
// One wave per edge; lane covers cols {lane, lane+32} for coalesced W reads.
// ---------------------------------------------------------------------------
__global__ __launch_bounds__(256)
void rgcn_edge_fused(const float* __restrict__ X,
                     const float* __restrict__ W,
                     const float* __restrict__ A,
                     const int* __restrict__ src,
                     const int* __restrict__ dst,
                     const int* __restrict__ etype,
                     float* __restrict__ Y,
                     int nEdges) {
    const int lane = threadIdx.x & 31;
    const long wave0  = (long)blockIdx.x * (blockDim.x >> 5) + (threadIdx.x >> 5);
    const long nWaves = (long)gridDim.x * (blockDim.x >> 5);
    for (long e = wave0; e < nEdges; e += nWaves) {
        int   s  = src[e];
        int   d  = dst[e];
        int   et = etype[e];
        float w  = A[e];
        const float* xrow = X + (size_t)s * F_DIM;
        const float* Wr   = W + (size_t)et * F_DIM * F_DIM;
        float acc0 = 0.0f, acc1 = 0.0f;
#pragma unroll 8
        for (int f = 0; f < F_DIM; ++f) {
            float xv = xrow[f];
            const float* wrow = Wr + (size_t)f * F_DIM;
            acc0 += xv * wrow[lane];
            acc1 += xv * wrow[lane + 32];
        }
        float* y = Y + (size_t)d * F_DIM;
        gadd_f32(y + lane,      acc0 * w);
        gadd_f32(y + lane + 32, acc1 * w);
    }
}

// ---------------------------------------------------------------------------
// Host launcher. Inputs (setup_inputs order): X, W, A, src, dst, etype.
// ---------------------------------------------------------------------------
extern "C" void kernel_launch(void* const* d_in, const int* in_sizes, int n_in,
                              void* d_out, int out_size, void* d_ws, size_t ws_size,
                              hipStream_t stream) {
    const float* X     = (const float*)d_in[0];
    const float* W     = (const float*)d_in[1];
    const float* A     = (const float*)d_in[2];
    const int*   src   = (const int*)d_in[3];
    const int*   dst   = (const int*)d_in[4];
    const int*   etype = (const int*)d_in[5];
    float*       Y     = (float*)d_out;

    const int nNodes = in_sizes[0] / F_DIM;
    const int nRels  = in_sizes[1] / (F_DIM * F_DIM);
    const int nEdges = in_sizes[2];

    // Zero the output accumulator.
    zero_kernel<<<2048, 256, 0, stream>>>(Y, (long)out_size);

    const size_t xwBytes = (size_t)nRels * (size_t)nNodes * F_DIM * sizeof(float);
    if (ws_size >= xwBytes) {
        float* XW = (float*)d_ws;
        const int rowTiles = (nNodes + 15) / 16;
        rgcn_xw_wmma<<<rowTiles, 32, 0, stream>>>(X, W, XW, nNodes, nRels);
        rgcn_edge_scatter<<<8192, 256, 0, stream>>>(XW, A, src, dst, etype, Y,
                                                    nEdges, nNodes);
    } else {
        rgcn_edge_fused<<<8192, 256, 0, stream>>>(X, W, A, src, dst, etype, Y,
                                                  nEdges);
    }
}